// GCN_23021024706912
// MI455X (gfx1250) — compile-verified
//
#include <hip/hip_runtime.h>

typedef __attribute__((ext_vector_type(2))) float v2f;
typedef __attribute__((ext_vector_type(8))) float v8f;

#define N_NODES 10000
#define N_EDGES 200000
#define D_IN    384
#define D_HID   48

// ---------------- degree / normalization ----------------

__global__ void init_deg_kernel(float* __restrict__ deg) {
    int i = blockIdx.x * blockDim.x + threadIdx.x;
    if (i < N_NODES) deg[i] = 1.0f;   // self-loop contribution
}

__global__ void deg_edges_kernel(const int* __restrict__ dst, float* __restrict__ deg) {
    int e = blockIdx.x * blockDim.x + threadIdx.x;
    if (e < N_EDGES) atomicAdd(&deg[dst[e]], 1.0f);
}

__global__ void dinv_kernel(float* __restrict__ deg) {
    int i = blockIdx.x * blockDim.x + threadIdx.x;
    if (i < N_NODES) {
        float d = deg[i];
        deg[i] = (d > 0.0f) ? rsqrtf(d) : 0.0f;   // deg >= 1 always, kept for fidelity
    }
}

// ---------------- dense GEMM via f32 WMMA ----------------
// Y[M,N] = X[M,K] @ W[N,K]^T (+bias). One wave per 16x16 tile of Y.
// f32 16x16x4 fragment layout (ISA 7.12.2):
//   A: lanes 0-15 M=lane, VGPR0/1 = K=(k0,k0+1); lanes 16-31 M=lane-16, K=(k0+2,k0+3)
//   B: same pattern with N in place of M (B[k][n] = W[n][k])
//   C/D: VGPR r -> M = r + 8*(lane>=16), N = lane&15
template<int N, int K, bool ADD_BIAS>
__global__ void gemm_xwT_wmma(const float* __restrict__ X,
                              const float* __restrict__ W,
                              const float* __restrict__ bias,
                              float* __restrict__ Y) {
    const int wave = (blockIdx.x * blockDim.x + threadIdx.x) >> 5;
    const int lane = threadIdx.x & 31;
    constexpr int TN = N / 16;
    const int tm = wave / TN;
    const int tn = wave % TN;
    if (tm >= N_NODES / 16) return;   // wave-uniform exit: EXEC stays all-1s

    const int lo = lane & 15;
    const int hi = lane >> 4;
    const float* __restrict__ arow = X + (size_t)(tm * 16 + lo) * K;
    const float* __restrict__ brow = W + (size_t)(tn * 16 + lo) * K;

    v8f c = {0.f, 0.f, 0.f, 0.f, 0.f, 0.f, 0.f, 0.f};
#pragma unroll 8
    for (int k0 = 0; k0 < K; k0 += 4) {
        const int ka = k0 + hi * 2;                 // even -> 8B aligned
        v2f a = *(const v2f*)(arow + ka);
        v2f b = *(const v2f*)(brow + ka);
        c = __builtin_amdgcn_wmma_f32_16x16x4_f32(
                /*neg_a=*/false, a, /*neg_b=*/false, b,
                /*c_mod=*/(short)0, c, /*reuse_a=*/false, /*reuse_b=*/false);
    }

#pragma unroll
    for (int r = 0; r < 8; ++r) {
        const int m = tm * 16 + r + hi * 8;
        const int n = tn * 16 + lo;
        float v = c[r];
        if (ADD_BIAS) v += bias[n];
        Y[(size_t)m * N + n] = v;
    }
}

// ---------------- aggregation (48-wide) ----------------

// OUT[i,:] = H[i,:] * dinv[i]^2   (self-loop term, also initializes OUT)
__global__ void selfloop_kernel(const float* __restrict__ H,
                                const float* __restrict__ dinv,
                                float* __restrict__ OUT) {
    int t = blockIdx.x * blockDim.x + threadIdx.x;
    if (t < N_NODES * D_HID) {
        float di = dinv[t / D_HID];
        OUT[t] = H[t] * di * di;
    }
}

// scatter-add: OUT[dst,:] += H[src,:] * dinv[src]*dinv[dst]
// one thread per (edge, 4-feature group); 48/4 = 12 groups per edge
__global__ void edge_agg_kernel(const float* __restrict__ H,
                                const int* __restrict__ src,
                                const int* __restrict__ dst,
                                const float* __restrict__ dinv,
                                float* __restrict__ OUT) {
    int t = blockIdx.x * blockDim.x + threadIdx.x;
    int e = t / 12;
    int g = t % 12;
    if (e >= N_EDGES) return;
    int s = src[e];
    int d = dst[e];
    float nrm = dinv[s] * dinv[d];
    float4 h4 = *(const float4*)(H + (size_t)s * D_HID + g * 4);  // L2-resident gather
    float* o = OUT + (size_t)d * D_HID + g * 4;
    atomicAdd(o + 0, h4.x * nrm);
    atomicAdd(o + 1, h4.y * nrm);
    atomicAdd(o + 2, h4.z * nrm);
    atomicAdd(o + 3, h4.w * nrm);
}

__global__ void bias_relu_kernel(float* __restrict__ Z, const float* __restrict__ b) {
    int t = blockIdx.x * blockDim.x + threadIdx.x;
    if (t < N_NODES * D_HID) {
        Z[t] = fmaxf(Z[t] + b[t % D_HID], 0.0f);
    }
}

// ---------------- launch ----------------

extern "C" void kernel_launch(void* const* d_in, const int* in_sizes, int n_in,
                              void* d_out, int out_size, void* d_ws, size_t ws_size,
                              hipStream_t stream) {
    const float* x  = (const float*)d_in[0];
    const int*   ei = (const int*)d_in[1];   // [2, E]: row0 = src, row1 = dst
    const float* W1 = (const float*)d_in[2];
    const float* b1 = (const float*)d_in[3];
    const float* W2 = (const float*)d_in[4];
    const float* b2 = (const float*)d_in[5];
    float* out = (float*)d_out;

    const int* src = ei;
    const int* dst = ei + N_EDGES;

    float* ws   = (float*)d_ws;
    float* dinv = ws;                         // 10000 (pad to 10240)
    float* h1   = ws + 10240;                 // 480000  x @ W1^T
    float* z    = h1 + N_NODES * D_HID;       // 480000  agg1 -> relu(+b1) in place
    float* agg2 = z  + N_NODES * D_HID;       // 480000  aggregated z

    const int B = 256;
    // degree -> 1/sqrt(deg)
    init_deg_kernel<<<(N_NODES + B - 1) / B, B, 0, stream>>>(dinv);
    deg_edges_kernel<<<(N_EDGES + B - 1) / B, B, 0, stream>>>(dst, dinv);
    dinv_kernel<<<(N_NODES + B - 1) / B, B, 0, stream>>>(dinv);

    // layer 1: h1 = x @ W1^T  (625*3 = 1875 tiles, 8 waves/block)
    {
        int waves  = (N_NODES / 16) * (D_HID / 16);
        int blocks = (waves + 7) / 8;
        gemm_xwT_wmma<D_HID, D_IN, false><<<blocks, B, 0, stream>>>(x, W1, nullptr, h1);
    }
    selfloop_kernel<<<(N_NODES * D_HID + B - 1) / B, B, 0, stream>>>(h1, dinv, z);
    edge_agg_kernel<<<(N_EDGES * 12 + B - 1) / B, B, 0, stream>>>(h1, src, dst, dinv, z);
    bias_relu_kernel<<<(N_NODES * D_HID + B - 1) / B, B, 0, stream>>>(z, b1);

    // layer 2: aggregate in 48-dim first (8x fewer atomics), then expand with W2
    selfloop_kernel<<<(N_NODES * D_HID + B - 1) / B, B, 0, stream>>>(z, dinv, agg2);
    edge_agg_kernel<<<(N_EDGES * 12 + B - 1) / B, B, 0, stream>>>(z, src, dst, dinv, agg2);

    // out = agg2 @ W2^T + b2  (625*24 = 15000 tiles)
    {
        int waves  = (N_NODES / 16) * (D_IN / 16);
        int blocks = (waves + 7) / 8;
        gemm_xwT_wmma<D_IN, D_HID, true><<<blocks, B, 0, stream>>>(agg2, W2, b2, out);
    }
}